// Attn_Pred_Model_16887811408123
// MI455X (gfx1250) — compile-verified
//
#include <hip/hip_runtime.h>

typedef float v2f __attribute__((ext_vector_type(2)));
typedef float v4f __attribute__((ext_vector_type(4)));
typedef float v8f __attribute__((ext_vector_type(8)));
typedef int   v4i __attribute__((ext_vector_type(4)));
typedef int   v8i __attribute__((ext_vector_type(8)));
typedef unsigned int v4u __attribute__((ext_vector_type(4)));

typedef __attribute__((address_space(1))) v4i gv4i;   // global int4
typedef __attribute__((address_space(3))) v4i sv4i;   // LDS int4
typedef __attribute__((address_space(3))) void lvoid; // LDS void

#define S_TOTAL   1024
#define D_CH      32
#define P_STEPS   128
#define S_BLK     128
#define HALO      128
#define TILE_ROWS (S_BLK + HALO)          // 256 rows of x staged in LDS
#define NKB       ((HALO + 16) / 4)       // 36 k-steps of 4 per 16x16 tile
#define WPAD_OFF  16
#define WPAD_SZ   160                     // zero-padded decay-weight table

#if defined(__has_builtin)
#  if __has_builtin(__builtin_amdgcn_tensor_load_to_lds)
#    define HAVE_TDM 1
#  endif
#  if __has_builtin(__builtin_amdgcn_global_load_async_to_lds_b128)
#    define HAVE_ASYNC_LDS 1
#  endif
#endif

// out[b,s,d] = sum_{p=0..127} alpha*beta^p * x[b,s-1-p,d]
//            + pos_fwd[d] + pos_bwd[((s - 32 d) & 1023) >> 5]
__global__ __launch_bounds__(128) void decay_attn_wmma(
    const float* __restrict__ x,
    const float* __restrict__ alpha_p,
    const float* __restrict__ beta_p,
    const float* __restrict__ pos_fwd,
    const float* __restrict__ pos_bwd,
    float* __restrict__ out)
{
    __shared__ float xs[TILE_ROWS * D_CH];   // 32 KB x tile (contiguous in memory)
    __shared__ float wpad[WPAD_SZ];          // alpha*beta^p, zero-padded band
    __shared__ float pfs[D_CH];
    __shared__ float pbs[D_CH];

    const int tid  = threadIdx.x;
    const int lane = tid & 31;
    const int wave = tid >> 5;               // 0..3
    const int blk  = blockIdx.x;             // 0..8191
    const int b    = blk >> 3;               // batch index
    const int sb   = (blk & 7) * S_BLK;      // s-block start

    const float alpha = alpha_p[0];
    const float beta  = beta_p[0];

    // Zero-padded weight table: wpad[WPAD_OFF + p] = alpha*beta^p, else 0.
    for (int i = tid; i < WPAD_SZ; i += 128) {
        const int p = i - WPAD_OFF;
        wpad[i] = (p >= 0 && p < P_STEPS) ? alpha * __powf(beta, (float)p) : 0.f;
    }
    if (tid < D_CH) { pfs[tid] = pos_fwd[tid]; pbs[tid] = pos_bwd[tid]; }

    // ---- Stage x[b, sb-128 .. sb+127, 0:32] -> LDS (contiguous 32 KB span;
    //      halo rows before s=0 are zero-filled; only the first s-block). ----
    const bool first = (sb == 0);
#if defined(HAVE_TDM)
    if (first) {                              // zero the 16 KB halo half
        v4f z; z.x = 0.f; z.y = 0.f; z.z = 0.f; z.w = 0.f;
        for (int i = tid; i < (HALO * D_CH) / 4; i += 128)
            *((v4f*)xs + i) = z;
    }
    if (wave == 0) {
        // One TDM descriptor: 1-D contiguous tile of n fp32 elements.
        const unsigned n =
            first ? (unsigned)(S_BLK * D_CH) : (unsigned)(TILE_ROWS * D_CH);
        const unsigned lds_dst =
            (unsigned)(unsigned long long)(lvoid*)xs + (first ? HALO * D_CH * 4u : 0u);
        const unsigned long long ga = (unsigned long long)
            (x + ((long)b * S_TOTAL + (first ? 0 : sb - HALO)) * D_CH);
        v4u g0;
        g0.x = 1u;                                        // count=1 valid descriptor
        g0.y = lds_dst;                                   // lds_addr (bytes)
        g0.z = (unsigned)ga;                              // global_addr[31:0]
        g0.w = ((unsigned)(ga >> 32) & 0x01FFFFFFu)       // global_addr[56:32]
             | 0x80000000u;                               // type=2 ("image")
        v8i g1;
        g1[0] = 0x00020000;                               // data_size=2 (4 bytes)
        g1[1] = (int)((n & 0xFFFFu) << 16);               // tensor_dim0[15:0]
        g1[2] = (int)((n >> 16) & 0xFFFFu) | (1 << 16);   // tensor_dim0[31:16], tensor_dim1=1
        g1[3] = (int)(n << 16);                           // tile_dim0 = n (fits 16 bits)
        g1[4] = 1;                                        // tile_dim1=1, tile_dim2=0
        g1[5] = (int)n;                                   // tensor_dim0_stride[31:0]
        g1[6] = 0;
        g1[7] = 0;
        v4i g2; g2.x = 0; g2.y = 0; g2.z = 0; g2.w = 0;   // unused (<=2-D tensor)
        v4i g3; g3.x = 0; g3.y = 0; g3.z = 0; g3.w = 0;
        v8i g4; g4[0] = 0; g4[1] = 0; g4[2] = 0; g4[3] = 0;
                g4[4] = 0; g4[5] = 0; g4[6] = 0; g4[7] = 0;
        __builtin_amdgcn_tensor_load_to_lds(g0, g1, g2, g3, g4, 0);
#  if __has_builtin(__builtin_amdgcn_s_wait_tensorcnt)
        __builtin_amdgcn_s_wait_tensorcnt(0);
#  else
        asm volatile("s_wait_tensorcnt 0x0" ::: "memory");
#  endif
    }
#else
    const long base = ((long)b * S_TOTAL + (long)(sb - HALO)) * D_CH;
    for (int i = tid; i < (TILE_ROWS * D_CH) / 4; i += 128) {
        if (first && (i >> 3) < HALO) {           // 8 float4 per 32-wide row
            v4f z; z.x = 0.f; z.y = 0.f; z.z = 0.f; z.w = 0.f;
            *((v4f*)xs + i) = z;
        } else {
#  if defined(HAVE_ASYNC_LDS)
            __builtin_amdgcn_global_load_async_to_lds_b128(
                (gv4i*)(x + base + (long)i * 4), (sv4i*)(xs + i * 4), 0, 0);
#  else
            *((v4f*)xs + i) = *(const v4f*)(x + base + (long)i * 4);
#  endif
        }
    }
#  if defined(HAVE_ASYNC_LDS)
#    if __has_builtin(__builtin_amdgcn_s_wait_asynccnt)
    __builtin_amdgcn_s_wait_asynccnt(0);
#    else
    asm volatile("s_wait_asynccnt 0x0" ::: "memory");
#    endif
#  endif
#endif
    __syncthreads();

    const int i16   = lane & 15;             // A row M / B col N / C col N
    const int khalf = (lane >> 4) * 2;       // K pair: lanes 0-15 -> {0,1}, 16-31 -> {2,3}
    const int mbase = (lane >> 4) * 8;       // C/D: lanes 16-31 hold M = 8..15

    // Hoist all 36 A-fragments (banded Toeplitz decay weights) into registers.
    // delta = s_out - s_in = i16 + HALO - (kb*4 + khalf); weight = wpad[delta-1+OFF].
    v2f afrag[NKB];
    #pragma unroll
    for (int kb = 0; kb < NKB; ++kb) {
        const int idx = i16 + HALO - (kb * 4 + khalf) - 1 + WPAD_OFF;
        v2f a;
        a.x = wpad[idx];        // K = khalf
        a.y = wpad[idx - 1];    // K = khalf+1  (delta one smaller)
        afrag[kb] = a;
    }

    // Each wave: 2 s-tiles, both d-tiles kept live (2 accumulators) so every
    // B row-pair feeds two WMMAs; B loads pipelined one k-step ahead.
    #pragma unroll
    for (int t = 0; t < 2; ++t) {
        const int srel = (wave * 2 + t) * 16;          // tile start within block
        v8f acc0 = {};                                 // d columns 0..15
        v8f acc1 = {};                                 // d columns 16..31
        const int r0 = srel + khalf;
        v2f b0, b1;
        b0.x = xs[r0 * D_CH + i16];       b0.y = xs[(r0 + 1) * D_CH + i16];
        b1.x = xs[r0 * D_CH + 16 + i16];  b1.y = xs[(r0 + 1) * D_CH + 16 + i16];
        #pragma unroll
        for (int kb = 0; kb < NKB; ++kb) {
            v2f nb0, nb1;
            if (kb + 1 < NKB) {                        // prefetch next B pair
                const int rn = srel + (kb + 1) * 4 + khalf;
                nb0.x = xs[rn * D_CH + i16];      nb0.y = xs[(rn + 1) * D_CH + i16];
                nb1.x = xs[rn * D_CH + 16 + i16]; nb1.y = xs[(rn + 1) * D_CH + 16 + i16];
            }
            acc0 = __builtin_amdgcn_wmma_f32_16x16x4_f32(
                false, afrag[kb], false, b0, (short)0, acc0, false, false);
            acc1 = __builtin_amdgcn_wmma_f32_16x16x4_f32(
                false, afrag[kb], false, b1, (short)0, acc1, false, false);
            if (kb + 1 < NKB) { b0 = nb0; b1 = nb1; }
        }
        // Store both 16x16 tiles. The pos_bwd bucket is provably constant
        // across the 8 rows a lane owns (sbase%8==0, never crosses a 32-run),
        // so the whole bias is one scalar per (tile, d-tile).
        const int sbase = sb + srel + mbase;
        #pragma unroll
        for (int dt = 0; dt < 2; ++dt) {
            const int   dcol = dt * 16 + i16;
            const int   bidx = ((sbase - 32 * dcol) & (S_TOTAL - 1)) >> 5;
            const float bias = pfs[dcol] + pbs[bidx];
            const v8f&  acc  = dt ? acc1 : acc0;
            float* op = out + ((long)b * S_TOTAL + sbase) * D_CH + dcol;
            #pragma unroll
            for (int v = 0; v < 8; ++v)
                op[v * D_CH] = acc[v] + bias;
        }
    }
}

extern "C" void kernel_launch(void* const* d_in, const int* in_sizes, int n_in,
                              void* d_out, int out_size, void* d_ws, size_t ws_size,
                              hipStream_t stream) {
    const float* x     = (const float*)d_in[0];
    const float* alpha = (const float*)d_in[1];
    const float* beta  = (const float*)d_in[2];
    const float* pf    = (const float*)d_in[3];
    const float* pb    = (const float*)d_in[4];
    float*       out   = (float*)d_out;

    const int B = 1024;
    dim3 grid(B * (S_TOTAL / S_BLK));   // 8192 workgroups
    dim3 block(128);                    // 4 waves (wave32)
    decay_attn_wmma<<<grid, block, 0, stream>>>(x, alpha, beta, pf, pb, out);
}